// Spiking1DLIFLayer_8246337208353
// MI455X (gfx1250) — compile-verified
//
#include <hip/hip_runtime.h>
#include <stdint.h>

// Problem shape (fixed by the reference): x[B, C, T] float32
#define BB 32
#define CC 512
#define TT 2048

// Tiling: one thread owns one (b,c) row; a wave owns 32 rows.
// Each TDM transfer stages a [32 rows x 64 floats] tile into LDS.
#define CHUNK          64                    // floats of T per chunk per row
#define NCHUNK         (TT / CHUNK)          // 32 sequential chunks per row
#define ROWS_PER_WAVE  32                    // wave32: one row per lane
#define WAVES_PER_BLK  4
#define ROWS_PER_BLK   (ROWS_PER_WAVE * WAVES_PER_BLK)   // 128
#define NBUF           3                     // triple buffer, prefetch distance 2

// TDM pads 4 dwords after every 64 dwords -> LDS row stride 68 dwords = 17 vec4
#define ROW_STRIDE_V4  17
#define BUF_V4         (ROWS_PER_WAVE * ROW_STRIDE_V4)   // 544 vec4 = 8704 B
#define BUF_BYTES      (BUF_V4 * 16)
#define SMEM_V4        (WAVES_PER_BLK * NBUF * BUF_V4)   // 104,448 B total LDS

typedef unsigned int v4u __attribute__((ext_vector_type(4)));
typedef int          v8i __attribute__((ext_vector_type(8)));
typedef int          v4i __attribute__((ext_vector_type(4)));
typedef float        v4f __attribute__((ext_vector_type(4)));

// Issue one TDM tensor load: 2D tile [ROWS_PER_WAVE x CHUNK] of f32,
// row stride T in memory, padded rows in LDS. D# bitfields per
// CDNA5 ISA (groups 0/1; groups 2/3 zero => 2D).
__device__ __forceinline__ void tdm_load_tile(const float* xg, uint64_t elt_off,
                                              uint32_t lds_byte)
{
  uint64_t addr = (uint64_t)(uintptr_t)xg + elt_off * 4ull;

  v4u g0;
  g0[0] = 1u;                                   // count=1 (valid user descriptor)
  g0[1] = lds_byte;                             // lds_addr (bytes; static LDS block @ offset 0)
  g0[2] = (uint32_t)addr;                       // global_addr[31:0]
  g0[3] = (uint32_t)((addr >> 32) & 0x1FFFFFFu) // global_addr[56:32]
        | (2u << 30);                           // type=2 ("image")

  v8i g1;
  g1[0] = (int)((2u << 16)     // data_size = 4 bytes
              | (1u << 20)     // pad_enable
              | (5u << 22)     // pad_interval: 64 dwords (256B)
              | (3u << 25));   // pad_amount:   4 dwords
  g1[1] = (int)(CHUNK << 16);          // [47:32]=atomic_barrier_addr=0 | tensor_dim0 lo16
  g1[2] = (int)(ROWS_PER_WAVE << 16);  // tensor_dim0 hi16=0 | tensor_dim1 lo16
  g1[3] = (int)(CHUNK << 16);          // tensor_dim1 hi16=0 | tile_dim0=64
  g1[4] = (int)(ROWS_PER_WAVE);        // tile_dim1=32 | tile_dim2=0 (2D)
  g1[5] = (int)(TT);                   // tensor_dim0_stride lo32 = T elements
  g1[6] = 0;                           // stride hi16 | tensor_dim1_stride lo16 (unused, 2D)
  g1[7] = 0;

  v4i z = {0, 0, 0, 0};
#if defined(__clang_major__) && (__clang_major__ >= 23)
  v8i z8 = {0, 0, 0, 0, 0, 0, 0, 0};
  __builtin_amdgcn_tensor_load_to_lds(g0, g1, z, z, z8, 0);
#else
  __builtin_amdgcn_tensor_load_to_lds(g0, g1, z, z, 0);
#endif
}

__global__ __launch_bounds__(ROWS_PER_BLK) void
lif_scan_kernel(const float* __restrict__ x, const float* __restrict__ Vth,
                const float* __restrict__ beta_p, float* __restrict__ out)
{
  // Only static LDS allocation in the kernel -> starts at LDS offset 0.
  __shared__ v4f smem[SMEM_V4];

  // The TDM engine writes smem, but the compiler only ever sees a numeric
  // LDS offset in the descriptor — smem looks like a never-written global and
  // loads from it would fold to undef. Escape its address into opaque asm so
  // every "memory" clobber below is assumed to (re)write the tile.
  void* lds_escape = (void*)smem;
  asm volatile("" : "+v"(lds_escape) :: "memory");

  const int lane = threadIdx.x & 31;
  const int wave = threadIdx.x >> 5;
  const int row       = blockIdx.x * ROWS_PER_BLK + wave * ROWS_PER_WAVE + lane;
  const int wave_row0 = blockIdx.x * ROWS_PER_BLK + wave * ROWS_PER_WAVE; // wave-uniform

  const float vth  = Vth[row & (CC - 1)];   // c = row % C (C power of two)
  const float beta = beta_p[0];

  const uint32_t bufsel   = (uint32_t)wave * NBUF;    // this wave's 3 buffers
  const uint64_t row_elt0 = (uint64_t)wave_row0 * TT;

  // Prime the pipeline: fetch chunks 0 and 1 (prefetch distance 2).
  tdm_load_tile(x, row_elt0,          (bufsel + 0u) * BUF_BYTES);
  tdm_load_tile(x, row_elt0 + CHUNK,  (bufsel + 1u) * BUF_BYTES);

  float mem = 0.0f, spk = 0.0f;
  float* orow = out + (size_t)row * TT;

  int cur = 0;                                   // k % NBUF, maintained incrementally
  for (int k = 0; k < NCHUNK; ++k) {
    if (k + 2 < NCHUNK) {
      // Refill the buffer this wave finished reading last iteration
      // (tile is wave-private: reads retired before this issue, no barrier).
      int nxt = cur + 2; if (nxt >= NBUF) nxt -= NBUF;
      tdm_load_tile(x, row_elt0 + (uint64_t)(k + 2) * CHUNK,
                    (bufsel + (uint32_t)nxt) * BUF_BYTES);
      __builtin_amdgcn_s_wait_tensorcnt(2);   // chunk k done; k+1, k+2 in flight
    } else if (k + 1 < NCHUNK) {
      __builtin_amdgcn_s_wait_tensorcnt(1);   // chunk k done; k+1 in flight
    } else {
      __builtin_amdgcn_s_wait_tensorcnt(0);   // last chunk done
    }
    // Renew the escape: LDS contents changed; keep ds reads below this point.
    asm volatile("" : "+v"(lds_escape) :: "memory");

    const uint32_t rbase = (bufsel + (uint32_t)cur) * BUF_V4
                         + (uint32_t)lane * ROW_STRIDE_V4;

#pragma unroll
    for (int q = 0; q < CHUNK / 4; ++q) {
      v4f xv = smem[rbase + (uint32_t)q];     // ds_load_b128, 16B aligned, bank-clean
      v4f sv;
      // LIF step x4: mem = (mem - spk*vth)*beta + x; spk = (mem > vth)
      mem = __builtin_fmaf(__builtin_fmaf(spk, -vth, mem), beta, xv.x);
      spk = (mem > vth) ? 1.0f : 0.0f;  sv.x = spk;
      mem = __builtin_fmaf(__builtin_fmaf(spk, -vth, mem), beta, xv.y);
      spk = (mem > vth) ? 1.0f : 0.0f;  sv.y = spk;
      mem = __builtin_fmaf(__builtin_fmaf(spk, -vth, mem), beta, xv.z);
      spk = (mem > vth) ? 1.0f : 0.0f;  sv.z = spk;
      mem = __builtin_fmaf(__builtin_fmaf(spk, -vth, mem), beta, xv.w);
      spk = (mem > vth) ? 1.0f : 0.0f;  sv.w = spk;
      // Spikes are write-once, never re-read: non-temporal b128 store.
      __builtin_nontemporal_store(sv, (v4f*)(orow + k * CHUNK) + q);
    }
    // Keep this iteration's ds reads above the next iteration's TDM issue.
    asm volatile("" : "+v"(lds_escape) :: "memory");

    ++cur; if (cur == NBUF) cur = 0;
  }
}

extern "C" void kernel_launch(void* const* d_in, const int* in_sizes, int n_in,
                              void* d_out, int out_size, void* d_ws, size_t ws_size,
                              hipStream_t stream)
{
  const float* x    = (const float*)d_in[0];  // [B, C, T] f32
  const float* vth  = (const float*)d_in[1];  // [C] f32
  const float* beta = (const float*)d_in[2];  // scalar f32
  float* out = (float*)d_out;                 // [B, C, T] f32 spikes

  dim3 grid((BB * CC) / ROWS_PER_BLK);        // 128 blocks
  dim3 block(ROWS_PER_BLK);                   // 128 threads = 4 wave32
  hipLaunchKernelGGL(lif_scan_kernel, grid, block, 0, stream, x, vth, beta, out);
}